// PhongBase_13829794693657
// MI455X (gfx1250) — compile-verified
//
#include <hip/hip_runtime.h>
#include <stdint.h>

// ---------------------------------------------------------------------------
// Phong streaming kernel for MI455X (gfx1250).
//   out[p,c] = ks[c]/pi + (kd[c]/pi) * max(dot(light_p, normal_p), 0)
// (pow(1,alpha)==1, so alpha is irrelevant.)
//
// Roofline: ~11 flops vs ~48 B per point (0.23 flop/B) -> pure HBM bandwidth,
// ~400 MB @ 23.3 TB/s ~= 17 us. WMMA has no role (no operand reuse).
// Data path: CDNA5 async global->LDS b128 copies (ASYNCcnt) stage the
// 36 B/point AoS records as fully-coalesced contiguous streams; compute reads
// LDS at a 9-dword stride (coprime with 64 banks -> conflict-free) and emits
// 12 B/point coalesced stores.
// ---------------------------------------------------------------------------

#define THREADS          256
#define TILE_PTS         1024
#define TILE_BYTES       (TILE_PTS * 36)                   // 36 KB LDS tile
#define LOADS_PER_THREAD (TILE_BYTES / (THREADS * 16))     // 9 x b128
#define PTS_PER_THREAD   (TILE_PTS / THREADS)              // 4

#if __has_builtin(__builtin_amdgcn_global_load_async_to_lds_b128)
#define HAVE_ASYNC_COPY 1
#else
#define HAVE_ASYNC_COPY 0
#endif

// The builtin's parameters are pointers to a 16-byte int vector in the
// global (AS1) and LDS (AS3) address spaces.
typedef int v4i __attribute__((vector_size(4 * sizeof(int))));
typedef __attribute__((address_space(1))) v4i* gptr_v4i;
typedef __attribute__((address_space(3))) v4i* lptr_v4i;

__device__ __forceinline__ gptr_v4i to_global_v4i(const void* p) {
  return (gptr_v4i)(uintptr_t)p;
}
__device__ __forceinline__ lptr_v4i to_lds_v4i(void* p) {
  // generic LDS pointer: low 32 bits are the LDS byte offset
  return (lptr_v4i)(uint32_t)(uintptr_t)p;
}

__device__ __forceinline__ void wait_async_zero() {
#if __has_builtin(__builtin_amdgcn_s_wait_asynccnt)
  __builtin_amdgcn_s_wait_asynccnt(0);
#else
  asm volatile("s_wait_asynccnt 0" ::: "memory");
#endif
}

__global__ __launch_bounds__(THREADS) void phong_tile_kernel(
    const float* __restrict__ in, const float* __restrict__ kd,
    const float* __restrict__ ks, float* __restrict__ out)
{
  __shared__ float tile[TILE_BYTES / 4];   // 9216 floats = 36 KB

  const int  t    = threadIdx.x;
  const char* src = (const char*)in + (unsigned long long)blockIdx.x * TILE_BYTES;

#if HAVE_ASYNC_COPY
#pragma unroll
  for (int i = 0; i < LOADS_PER_THREAD; ++i) {
    const int off = i * (THREADS * 16) + t * 16;
    __builtin_amdgcn_global_load_async_to_lds_b128(
        to_global_v4i(src + off), to_lds_v4i((char*)tile + off), 0, 0);
  }
#else
#pragma unroll
  for (int i = 0; i < LOADS_PER_THREAD; ++i) {
    const int off = i * (THREADS * 16) + t * 16;
    *(float4*)((char*)tile + off) = *(const float4*)(src + off);
  }
#endif

  // Uniform coefficient loads (s_load) overlap the in-flight DMA.
  const float inv_pi = 0.31830988618379067f;
  const float a0 = ks[0] * inv_pi, a1 = ks[1] * inv_pi, a2 = ks[2] * inv_pi;
  const float b0 = kd[0] * inv_pi, b1 = kd[1] * inv_pi, b2 = kd[2] * inv_pi;

#if HAVE_ASYNC_COPY
  wait_async_zero();
#endif
  __syncthreads();

  float* obase = out + (unsigned long long)blockIdx.x * (TILE_PTS * 3);
#pragma unroll
  for (int k = 0; k < PTS_PER_THREAD; ++k) {
    const int p = k * THREADS + t;                 // tile-local point
    const float* r = &tile[p * 9];                 // stride-9 dwords: conflict-free
    const float dot = r[0] * r[3] + r[1] * r[4] + r[2] * r[5];
    const float nd  = fmaxf(dot, 0.0f);
    float* o = obase + p * 3;
    o[0] = a0 + b0 * nd;
    o[1] = a1 + b1 * nd;
    o[2] = a2 + b2 * nd;
  }
}

// Generic tail for npts not a multiple of TILE_PTS (not hit for N = 8388608).
__global__ void phong_tail_kernel(
    const float* __restrict__ in, const float* __restrict__ kd,
    const float* __restrict__ ks, float* __restrict__ out,
    int start, int npts)
{
  const int p = start + blockIdx.x * blockDim.x + threadIdx.x;
  if (p >= npts) return;
  const float inv_pi = 0.31830988618379067f;
  const float* r = in + (unsigned long long)p * 9;
  const float dot = r[0] * r[3] + r[1] * r[4] + r[2] * r[5];
  const float nd  = fmaxf(dot, 0.0f);
  float* o = out + (unsigned long long)p * 3;
  o[0] = ks[0] * inv_pi + kd[0] * inv_pi * nd;
  o[1] = ks[1] * inv_pi + kd[1] * inv_pi * nd;
  o[2] = ks[2] * inv_pi + kd[2] * inv_pi * nd;
}

extern "C" void kernel_launch(void* const* d_in, const int* in_sizes, int n_in,
                              void* d_out, int out_size, void* d_ws, size_t ws_size,
                              hipStream_t stream) {
  const float* in = (const float*)d_in[0];   // (N,3,3) f32
  const float* kd = (const float*)d_in[1];   // (3,)   f32
  const float* ks = (const float*)d_in[2];   // (3,)   f32
  // d_in[3] = alpha: pow(1.0, alpha) == 1.0, unused.
  float* out = (float*)d_out;                // (N,3)  f32

  const long long npts   = (long long)in_sizes[0] / 9;
  const long long ntiles = npts / TILE_PTS;

  if (ntiles > 0) {
    phong_tile_kernel<<<(unsigned)ntiles, THREADS, 0, stream>>>(in, kd, ks, out);
  }
  const long long start = ntiles * TILE_PTS;
  const long long rem   = npts - start;
  if (rem > 0) {
    const int blocks = (int)((rem + THREADS - 1) / THREADS);
    phong_tail_kernel<<<blocks, THREADS, 0, stream>>>(in, kd, ks, out,
                                                      (int)start, (int)npts);
  }
}